// Environment_33105607918121
// MI455X (gfx1250) — compile-verified
//
#include <hip/hip_runtime.h>
#include <cstddef>

typedef __attribute__((ext_vector_type(2))) float v2f;
typedef __attribute__((ext_vector_type(8))) float v8f;

__device__ __forceinline__ float sigmoidf(float v) {
    return 1.0f / (1.0f + expf(-v));
}

// ---------------------------------------------------------------------------
// Bulk float4 copy (bandwidth-bound part: ~290 MB read + 246 MB write total)
// ---------------------------------------------------------------------------
__global__ void copy_f4(const float4* __restrict__ src, float4* __restrict__ dst,
                        long n4) {
    long i = (long)blockIdx.x * blockDim.x + threadIdx.x;
    long stride = (long)gridDim.x * blockDim.x;
    for (; i < n4; i += stride) dst[i] = src[i];
}

__global__ void zero_i32(int* __restrict__ p, int n) {
    int i = blockIdx.x * blockDim.x + threadIdx.x;
    if (i < n) p[i] = 0;
}

// ---------------------------------------------------------------------------
// GEMV y[0..M) = A(MxK, row-major) @ w(K)  via V_WMMA_F32_16X16X4_F32.
// One wave -> 16 rows. A-tile 16x4 f32 layout: lane L (L&15 = M-row),
// khalf = L>>4 selects K={0,1} vs K={2,3} in the two A VGPRs.
// B-tile = vector in column N==0 (nonzero only in lanes 0 and 16).
// D column 0 lives in lane 0 (M=0..7, VGPR 0..7) and lane 16 (M=8..15).
// Optionally fuses y = sigmoid(y + bias).
// ---------------------------------------------------------------------------
__global__ void gemv_wmma(const float* __restrict__ A, const float* __restrict__ w,
                          const float* __restrict__ bias, float* __restrict__ y,
                          int M, int K, int apply_sigmoid) {
    const int lane  = threadIdx.x & 31;
    const int wave  = threadIdx.x >> 5;
    const int row0  = (blockIdx.x * (blockDim.x >> 5) + wave) << 4;
    if (row0 >= M) return;                 // wave-uniform guard
    const int m     = lane & 15;           // row within tile / N within B row
    const int khalf = lane >> 4;           // 0 or 1

    const float* ap = A + (size_t)(row0 + m) * K + 2 * khalf;
    const float* wp = w + 2 * khalf;

    v8f c = {0.f, 0.f, 0.f, 0.f, 0.f, 0.f, 0.f, 0.f};
    for (int k0 = 0; k0 < K; k0 += 4) {
        v2f a  = *(const v2f*)(ap + k0);
        v2f wv = *(const v2f*)(wp + k0);
        v2f b;
        b.x = (m == 0) ? wv.x : 0.0f;      // B[k][0] = w[k], B[k][n>0] = 0
        b.y = (m == 0) ? wv.y : 0.0f;
        c = __builtin_amdgcn_wmma_f32_16x16x4_f32(
                /*neg_a=*/false, a, /*neg_b=*/false, b,
                /*c_mod=*/(short)0, c, /*reuse_a=*/false, /*reuse_b=*/false);
    }
    if (m == 0) {                          // lanes 0 and 16 hold column N=0
        const int base = row0 + khalf * 8;
        #pragma unroll
        for (int r = 0; r < 8; ++r) {
            float v = c[r];
            if (bias) v += bias[base + r];
            if (apply_sigmoid) v = sigmoidf(v);
            y[base + r] = v;
        }
    }
}

// ---------------------------------------------------------------------------
// Phase C: dP = x[POI].t_vec, dU = s_u[user].t_vec, then
//   out_su[user] = sigmoid(s_u[user] + W_u * dP)
//   new_POI      = sigmoid(x[POI]   + W_p * dU)   -> out_x[POI], ws
// ---------------------------------------------------------------------------
__global__ void phase_c(const float* __restrict__ s_u, const float* __restrict__ x,
                        const float* __restrict__ W_u, const float* __restrict__ W_p,
                        const float* __restrict__ tv, const int* __restrict__ puser,
                        const int* __restrict__ ppoi, float* __restrict__ out_su,
                        float* __restrict__ out_x, float* __restrict__ newPOI, int d) {
    __shared__ float redP[256];
    __shared__ float redU[256];
    const int tid  = threadIdx.x;
    const int uidx = *puser;
    const int pidx = *ppoi;
    const float* cu = s_u + (size_t)uidx * d;
    const float* cp = x   + (size_t)pidx * d;

    float sP = 0.f, sU = 0.f;
    for (int i = tid; i < d; i += 256) {
        float t = tv[i];
        sP += cp[i] * t;   // dot(cur_POI, t_vec)
        sU += cu[i] * t;   // dot(cur_user, t_vec)
    }
    redP[tid] = sP; redU[tid] = sU;
    __syncthreads();
    for (int s = 128; s > 0; s >>= 1) {
        if (tid < s) { redP[tid] += redP[tid + s]; redU[tid] += redU[tid + s]; }
        __syncthreads();
    }
    const float dP = redP[0];
    const float dU = redU[0];
    for (int i = tid; i < d; i += 256) {
        out_su[(size_t)uidx * d + i] = sigmoidf(cu[i] + W_u[i] * dP);
        float np = sigmoidf(cp[i] + W_p[i] * dU);
        out_x[(size_t)pidx * d + i] = np;
        newPOI[i] = np;
    }
}

// ---------------------------------------------------------------------------
// Tail scatter: edges with head == POI:  out_x[tail] = new_POI + edge_attr[e],
// and mark flags[tail] = 1 (tail-set for the neighbor mask).
// ---------------------------------------------------------------------------
__global__ void tail_update(const int* __restrict__ edge_index,
                            const float* __restrict__ edge_attr,
                            const float* __restrict__ newPOI,
                            const int* __restrict__ ppoi,
                            float* __restrict__ out_x, int* __restrict__ flags,
                            int E, int d) {
    const int e   = blockIdx.x;
    const int poi = *ppoi;
    if (edge_index[e] != poi) return;            // heads[e] == POI ?
    const int tail = edge_index[E + e];
    float* dst = out_x + (size_t)tail * d;
    const float* ea = edge_attr + (size_t)e * d;
    for (int i = threadIdx.x; i < d; i += 256) dst[i] = newPOI[i] + ea[i];
    if (threadIdx.x == 0) flags[tail] = 1;
}

// ---------------------------------------------------------------------------
// Neighbor update: for e != POI_index with tails[e] in tail-set:
//   scal = (x[head_e] - edge_attr[e]) . W_p_
//   out_x[head_e] = sigmoid(x[head_e] + scal)
// (reads the tail-updated out_x, as in the reference)
// ---------------------------------------------------------------------------
__global__ void neighbor_update(const int* __restrict__ edge_index,
                                const float* __restrict__ edge_attr,
                                const float* __restrict__ W_p_,
                                const int* __restrict__ ppoi,
                                float* __restrict__ out_x,
                                const int* __restrict__ flags, int E, int d) {
    __shared__ float red[256];
    const int e   = blockIdx.x;
    const int poi = *ppoi;
    if (e == poi) return;                        // neigh_mask[POI_index] = False
    const int tail = edge_index[E + e];
    if (!flags[tail]) return;
    const int head = edge_index[e];
    float* row = out_x + (size_t)head * d;
    const float* ea = edge_attr + (size_t)e * d;

    const int tid = threadIdx.x;
    float s = 0.f;
    for (int i = tid; i < d; i += 256) s += (row[i] - ea[i]) * W_p_[i];
    red[tid] = s;
    __syncthreads();
    for (int st = 128; st > 0; st >>= 1) {
        if (tid < st) red[tid] += red[tid + st];
        __syncthreads();
    }
    const float scal = red[0];
    for (int i = tid; i < d; i += 256) row[i] = sigmoidf(row[i] + scal);
}

// ---------------------------------------------------------------------------
extern "C" void kernel_launch(void* const* d_in, const int* in_sizes, int n_in,
                              void* d_out, int out_size, void* d_ws, size_t ws_size,
                              hipStream_t stream) {
    const float* s_u       = (const float*)d_in[0];
    const float* x         = (const float*)d_in[1];
    const float* edge_attr = (const float*)d_in[2];
    const float* T         = (const float*)d_in[3];
    const float* W_u       = (const float*)d_in[4];
    const float* W_p       = (const float*)d_in[5];
    const float* W_T_1     = (const float*)d_in[6];
    const float* W_T_2     = (const float*)d_in[7];
    const float* b_T       = (const float*)d_in[8];
    const float* W_p_      = (const float*)d_in[9];
    const int*   edge_idx  = (const int*)d_in[10];
    const int*   puser     = (const int*)d_in[11];
    const int*   ppoi      = (const int*)d_in[12];

    const int  d        = in_sizes[4];              // 1024 (W_u has d elems)
    const long su_elems = in_sizes[0];              // N_USERS * d
    const long x_elems  = in_sizes[1];              // N_NODES * d
    const int  n_nodes  = (int)(x_elems / d);
    const int  E        = in_sizes[10] / 2;         // 4096
    const int  n_t1     = in_sizes[6] / d;          // 2048 (W_T_1 is d x N_T1)
    const int  n_t2     = in_sizes[7];              // 2048 (W_T_2 is N_T2)

    float* out_su = (float*)d_out;
    float* out_x  = out_su + su_elems;

    // workspace layout
    float* ws_f   = (float*)d_ws;
    float* tmp_v  = ws_f;                 // n_t1 floats: T @ W_T_2
    float* tv     = tmp_v + n_t1;         // d floats: t_vec
    float* newPOI = tv + d;               // d floats
    int*   flags  = (int*)(newPOI + d);   // n_nodes ints

    // 1) bulk copies into d_out (dominant ~500 MB of HBM traffic)
    copy_f4<<<2048, 256, 0, stream>>>((const float4*)s_u, (float4*)out_su, su_elems / 4);
    copy_f4<<<2048, 256, 0, stream>>>((const float4*)x,   (float4*)out_x,  x_elems  / 4);

    // 2) clear tail-set flags
    zero_i32<<<(n_nodes + 255) / 256, 256, 0, stream>>>(flags, n_nodes);

    // 3) tmp = T @ W_T_2               (WMMA f32 16x16x4)
    gemv_wmma<<<(n_t1 + 127) / 128, 256, 0, stream>>>(T, W_T_2, nullptr, tmp_v,
                                                      n_t1, n_t2, 0);
    // 4) t_vec = sigmoid(W_T_1 @ tmp + b_T)
    gemv_wmma<<<(d + 127) / 128, 256, 0, stream>>>(W_T_1, tmp_v, b_T, tv,
                                                   d, n_t1, 1);
    // 5) user / POI row updates, new_POI to workspace
    phase_c<<<1, 256, 0, stream>>>(s_u, x, W_u, W_p, tv, puser, ppoi,
                                   out_su, out_x, newPOI, d);
    // 6) tail scatter + tail-set flags
    tail_update<<<E, 256, 0, stream>>>(edge_idx, edge_attr, newPOI, ppoi,
                                       out_x, flags, E, d);
    // 7) neighbor updates
    neighbor_update<<<E, 256, 0, stream>>>(edge_idx, edge_attr, W_p_, ppoi,
                                           out_x, flags, E, d);
}